// SI_37898791420506
// MI455X (gfx1250) — compile-verified
//
#include <hip/hip_runtime.h>
#include <hip/hip_bf16.h>

// Problem constants (reference: B=16, C=128, D=256)
#define NB 16
#define NC 128
#define ND 256
#define NROWS (NB * NC)      // 2048 flattened (b,row) rows

typedef __attribute__((ext_vector_type(2))) float v2f;
typedef __attribute__((ext_vector_type(8))) float v8f;

static __device__ __forceinline__ v8f wmma4(v2f a, v2f b, v8f c) {
    // D = A(16x4,f32) * B(4x16,f32) + C(16x16,f32), wave32
    return __builtin_amdgcn_wmma_f32_16x16x4_f32(
        /*neg_a=*/false, a, /*neg_b=*/false, b,
        /*c_mod=*/(short)0, c, /*reuse_a=*/false, /*reuse_b=*/false);
}

// ---------------------------------------------------------------------------
// Kernel 1: P = H @ W1[:D], Q = H @ W1[D:]   (H = h_t flattened (2048 x 256))
// One wave -> one 16x16 tile of BOTH P and Q (A-fragments shared).
// tiles: Mt=128, Nt=16 -> 2048 tiles, 8 waves/block -> 256 blocks.
// ---------------------------------------------------------------------------
__global__ void __launch_bounds__(256) k_gemm_pq(const float* __restrict__ H,
                                                 const float* __restrict__ W1,
                                                 float* __restrict__ P,
                                                 float* __restrict__ Q) {
    const int lane = threadIdx.x & 31;
    const int wv   = threadIdx.x >> 5;
    const int tile = blockIdx.x * 8 + wv;
    const int mt = tile >> 4;          // 0..127
    const int nt = tile & 15;          // 0..15
    const int hi = lane >> 4, lo = lane & 15;
    const int n  = nt * 16 + lo;

    const float* Arow = H + (mt * 16 + lo) * ND;
    v8f accP = {}, accQ = {};
#pragma unroll 4
    for (int k = 0; k < ND; k += 4) {
        const int ka = k + 2 * hi;                       // even -> 8B aligned
        v2f a = *(const v2f*)(Arow + ka);
        v2f bp, bq;
        bp.x = W1[ka * ND + n];
        bp.y = W1[(ka + 1) * ND + n];
        bq.x = W1[(ka + ND) * ND + n];
        bq.y = W1[(ka + ND + 1) * ND + n];
        accP = wmma4(a, bp, accP);
        accQ = wmma4(a, bq, accQ);
    }
#pragma unroll
    for (int g = 0; g < 8; ++g) {
        const int row = mt * 16 + g + 8 * hi;
        P[row * ND + n] = accP[g];
        Q[row * ND + n] = accQ[g];
    }
}

// ---------------------------------------------------------------------------
// Kernel 2: pairwise edge scores (non-linear, not a GEMM):
//   G[b,u,v] = sigmoid( b2 + sum_k relu(P[b,u,k]+Q[b,v,k]+b1[k]) * W2[k] ) * A_a[u,v]
// One block -> one (b, 16u x 16v) tile; P/Q tiles staged in LDS with stride
// 257 (odd) so the strided Q-column reads hit distinct banks.
// grid = 16 * 8 * 8 = 1024 blocks, 256 threads (one (u,v) pair per thread).
// ---------------------------------------------------------------------------
__global__ void __launch_bounds__(256) k_scores(const float* __restrict__ P,
                                                const float* __restrict__ Q,
                                                const float* __restrict__ b1,
                                                const float* __restrict__ W2,
                                                const float* __restrict__ b2,
                                                const float* __restrict__ A_a,
                                                float* __restrict__ G) {
    __shared__ float Pl[16][ND + 1];
    __shared__ float Ql[16][ND + 1];
    __shared__ float b1s[ND];
    __shared__ float w2s[ND];

    const int blk = blockIdx.x;
    const int vt = blk & 7;
    const int ut = (blk >> 3) & 7;
    const int b  = blk >> 6;
    const int t  = threadIdx.x;

#pragma unroll
    for (int i = t; i < 16 * ND; i += 256) {
        const int r = i >> 8, k = i & (ND - 1);
        Pl[r][k] = P[(b * NC + ut * 16 + r) * ND + k];
        Ql[r][k] = Q[(b * NC + vt * 16 + r) * ND + k];
    }
    b1s[t] = b1[t];
    w2s[t] = W2[t];
    __syncthreads();

    const int u = t >> 4, v = t & 15;
    float acc = 0.f;
#pragma unroll 8
    for (int k = 0; k < ND; ++k) {
        const float x = Pl[u][k] + Ql[v][k] + b1s[k];
        acc = fmaf(fmaxf(x, 0.f), w2s[k], acc);
    }
    const float s  = 1.0f / (1.0f + __expf(-(acc + b2[0])));
    const int uu = ut * 16 + u, vv = vt * 16 + v;
    G[(b * NC + uu) * NC + vv] = s * A_a[uu * NC + vv];
}

// ---------------------------------------------------------------------------
// Kernel 3: per-batch aggregations, fused (share the B-operand h):
//   a_t[b,v,d] = sum_u G[b,u,v]   * h[b,u,d]
//   m_t[b,v,d] = (sum_u A_m[u,v]  * h[b,u,d]) * h[b,v,d]
// A-matrix[m=v][k=u]; K=128. tiles: 16 batches * 8 vt * 16 dt = 2048 tiles.
// ---------------------------------------------------------------------------
__global__ void __launch_bounds__(256) k_agg(const float* __restrict__ G,
                                             const float* __restrict__ Am,
                                             const float* __restrict__ h,
                                             float* __restrict__ a_t,
                                             float* __restrict__ m_t) {
    const int lane = threadIdx.x & 31;
    const int wv   = threadIdx.x >> 5;
    const int tile = blockIdx.x * 8 + wv;
    const int dt = tile & 15;
    const int vt = (tile >> 4) & 7;
    const int b  = tile >> 7;
    const int hi = lane >> 4, lo = lane & 15;
    const int n  = dt * 16 + lo;   // d column
    const int m  = vt * 16 + lo;   // v row (column index into G / Am)

    const float* Gb = G + b * NC * NC;
    const float* hb = h + b * NC * ND;
    v8f accA = {}, accM = {};
#pragma unroll 4
    for (int k = 0; k < NC; k += 4) {
        const int ka = k + 2 * hi;
        v2f ag, am, bb;
        ag.x = Gb[ka * NC + m];
        ag.y = Gb[(ka + 1) * NC + m];
        am.x = Am[ka * NC + m];
        am.y = Am[(ka + 1) * NC + m];
        bb.x = hb[ka * ND + n];
        bb.y = hb[(ka + 1) * ND + n];
        accA = wmma4(ag, bb, accA);
        accM = wmma4(am, bb, accM);
    }
#pragma unroll
    for (int g = 0; g < 8; ++g) {
        const int row = vt * 16 + g + 8 * hi;
        a_t[(b * NC + row) * ND + n] = accA[g];
        m_t[(b * NC + row) * ND + n] = accM[g] * hb[row * ND + n];
    }
}

// ---------------------------------------------------------------------------
// Kernel 4: z = beta1*h + beta2*(a_t@Wa + ba) + beta3*(m_t@Wm + bm)
// Two 2048x256x256 GEMMs fused into one output tile.
// ---------------------------------------------------------------------------
__global__ void __launch_bounds__(256) k_final(const float* __restrict__ a_t,
                                               const float* __restrict__ m_t,
                                               const float* __restrict__ h,
                                               const float* __restrict__ Wa,
                                               const float* __restrict__ Wm,
                                               const float* __restrict__ ba,
                                               const float* __restrict__ bm,
                                               const float* __restrict__ pb1,
                                               const float* __restrict__ pb2,
                                               const float* __restrict__ pb3,
                                               float* __restrict__ z) {
    const int lane = threadIdx.x & 31;
    const int wv   = threadIdx.x >> 5;
    const int tile = blockIdx.x * 8 + wv;
    const int mt = tile >> 4;
    const int nt = tile & 15;
    const int hi = lane >> 4, lo = lane & 15;
    const int n  = nt * 16 + lo;

    const float* Ar = a_t + (mt * 16 + lo) * ND;
    const float* Mr = m_t + (mt * 16 + lo) * ND;
    v8f acc1 = {}, acc2 = {};
#pragma unroll 4
    for (int k = 0; k < ND; k += 4) {
        const int ka = k + 2 * hi;
        v2f a1 = *(const v2f*)(Ar + ka);
        v2f a2 = *(const v2f*)(Mr + ka);
        v2f w1v, w2v;
        w1v.x = Wa[ka * ND + n];
        w1v.y = Wa[(ka + 1) * ND + n];
        w2v.x = Wm[ka * ND + n];
        w2v.y = Wm[(ka + 1) * ND + n];
        acc1 = wmma4(a1, w1v, acc1);
        acc2 = wmma4(a2, w2v, acc2);
    }
    const float beta1 = pb1[0], beta2 = pb2[0], beta3 = pb3[0];
    const float bav = ba[n], bmv = bm[n];
#pragma unroll
    for (int g = 0; g < 8; ++g) {
        const int row = mt * 16 + g + 8 * hi;
        z[row * ND + n] = beta1 * h[row * ND + n]
                        + beta2 * (acc1[g] + bav)
                        + beta3 * (acc2[g] + bmv);
    }
}

// ---------------------------------------------------------------------------
extern "C" void kernel_launch(void* const* d_in, const int* in_sizes, int n_in,
                              void* d_out, int out_size, void* d_ws, size_t ws_size,
                              hipStream_t stream) {
    (void)in_sizes; (void)n_in; (void)out_size; (void)ws_size;

    const float* h_t = (const float*)d_in[0];   // (B,C,D)
    const float* A_a = (const float*)d_in[1];   // (C,C)
    const float* A_m = (const float*)d_in[2];   // (C,C)
    const float* W1  = (const float*)d_in[3];   // (2D,D)
    const float* b1  = (const float*)d_in[4];   // (D,)
    const float* W2  = (const float*)d_in[5];   // (D,1)
    const float* b2  = (const float*)d_in[6];   // (1,)
    const float* Wa  = (const float*)d_in[7];   // (D,D)
    const float* ba  = (const float*)d_in[8];   // (D,)
    const float* Wm  = (const float*)d_in[9];   // (D,D)
    const float* bm  = (const float*)d_in[10];  // (D,)
    const float* be1 = (const float*)d_in[11];
    const float* be2 = (const float*)d_in[12];
    const float* be3 = (const float*)d_in[13];
    float* z = (float*)d_out;

    // Workspace layout (floats): P, Q (2048x256 each), G (16x128x128),
    // a_t, m_t (2048x256 each)  -> ~9.4 MB total.
    float* P  = (float*)d_ws;
    float* Q  = P + NROWS * ND;
    float* G  = Q + NROWS * ND;
    float* At = G + NB * NC * NC;
    float* Mt = At + NROWS * ND;

    // 1) pre_u / pre_v : 2048 tiles, 8 waves/block
    k_gemm_pq<<<256, 256, 0, stream>>>(h_t, W1, P, Q);
    // 2) pairwise scores * A_a : one block per (b, u-tile, v-tile)
    k_scores<<<NB * 8 * 8, 256, 0, stream>>>(P, Q, b1, W2, b2, A_a, G);
    // 3) fused additive + hadamard aggregation
    k_agg<<<256, 256, 0, stream>>>(G, A_m, h_t, At, Mt);
    // 4) fused output GEMMs + epilogue
    k_final<<<256, 256, 0, stream>>>(At, Mt, h_t, Wa, Wm, ba, bm,
                                     be1, be2, be3, z);
}